// MetaMultiLinear_19602230739476
// MI455X (gfx1250) — compile-verified
//
#include <hip/hip_runtime.h>

#define H     8
#define BATCH 32768
#define IN_F  16
#define COND  32
#define OUT_F 32
#define KPI   17                         // IN_F + 1 (homogeneous column)
#define KTOT  544                        // OUT_F * KPI
#define WRSZ  (H * KPI * OUT_F * COND)   // 139264 floats reordered weights
#define BRSZ  (H * KPI * OUT_F)          // 4352 floats reordered bias
#define OSTR  36                         // LDS o-row stride in dwords (bank-conflict-free, 16B aligned)

typedef __attribute__((ext_vector_type(2))) float v2f;
typedef __attribute__((ext_vector_type(4))) float v4f;
typedef __attribute__((ext_vector_type(8))) float v8f;

// ---------------------------------------------------------------------------
// Kernel 1: reorder cond_weight (h, o*17+i, c) -> Wr (h, i, o, c) and
//           cond_bias (h, o*17+i) -> biasr (h, i, o).  ~574 KB of d_ws.
// ---------------------------------------------------------------------------
__global__ __launch_bounds__(256) void reorder_kernel(
    const float* __restrict__ W, const float* __restrict__ bias,
    float* __restrict__ Wr, float* __restrict__ biasr) {
  int tid = blockIdx.x * 256 + threadIdx.x;
  if (tid < WRSZ) {
    int h  = tid / (KPI * OUT_F * COND);
    int r1 = tid % (KPI * OUT_F * COND);
    int i  = r1 / (OUT_F * COND);
    int r2 = r1 % (OUT_F * COND);
    int o  = r2 >> 5;
    int c  = r2 & 31;
    Wr[tid] = W[(size_t)(h * KTOT + o * KPI + i) * COND + c];
  }
  if (tid < BRSZ) {
    int h = tid / (KPI * OUT_F);
    int r = tid % (KPI * OUT_F);
    int i = r / OUT_F;
    int o = r % OUT_F;
    biasr[tid] = bias[h * KTOT + o * KPI + i];
  }
}

// ---------------------------------------------------------------------------
// Kernel 2: fused hypernetwork GEMM + per-sample matvec.
// Block = 8 waves, all on the same head; head's reordered weight slab is
// staged once into bank-conflict-free padded LDS, then each wave runs one
// 16-sample tile: per i, 16x WMMA f32 16x16x4 (B from LDS) + fused
// (w + bias) * x1[m][i] accumulation into the output fragment.
// ---------------------------------------------------------------------------
__global__ __launch_bounds__(256) void meta_kernel(
    const float* __restrict__ input, const float* __restrict__ cond,
    const float* __restrict__ Wr, const float* __restrict__ biasr,
    float* __restrict__ out) {
  __shared__ float wLds[KPI * OUT_F * OSTR];   // 17*32*36*4 = 78336 B, padded [i][o]
  __shared__ float bLds[KPI * OUT_F];          // 2176 B
  __shared__ float x1s[8][16 * KPI];           // per-wave [m][i] tile, i==16 -> 1.0

  const int tid  = threadIdx.x;
  const int lane = tid & 31;
  const int wib  = tid >> 5;
  const int h    = blockIdx.x >> 8;            // 256 blocks per head
  const int tile = (blockIdx.x & 255) * 8 + wib;
  const int base = tile * 16;
  const int c16  = lane & 15;
  const int half = lane >> 4;

  // ---- cooperative stage: Wr[h] (17408 floats) -> padded LDS ----
  const v4f* src = (const v4f*)(Wr + (size_t)h * KPI * OUT_F * COND);
  #pragma unroll 1
  for (int t = 0; t < 17; ++t) {
    int s  = t * 256 + tid;                    // vec4 index, 4352 total (exact)
    v4f v  = src[s];
    int io = s >> 3;                           // (i*32 + o): 8 vec4 per 32-float row
    int c4 = (s & 7) * 4;
    *(v4f*)&wLds[io * OSTR + c4] = v;          // row start 144B-aligned, c4*4 mult of 16
  }
  // ---- stage bias (544 floats) ----
  #pragma unroll
  for (int t = 0; t < 3; ++t) {
    int idx = t * 256 + tid;
    if (idx < KPI * OUT_F) bLds[idx] = biasr[h * KPI * OUT_F + idx];
  }
  // ---- stage x1 = [input_tile | 1] (272 floats per wave) ----
  const float* inpH = input + (size_t)(h * BATCH + base) * IN_F;
  #pragma unroll
  for (int t = 0; t < 9; ++t) {
    int idx = t * 32 + lane;
    if (idx < 16 * KPI) {
      int m = idx / KPI;
      int i = idx - m * KPI;
      x1s[wib][idx] = (i < IN_F) ? inpH[m * IN_F + i] : 1.0f;
    }
  }

  // ---- load A fragments: cond tile, 8 fragments of K=4 (fp32 WMMA layout) ----
  const float* condH = cond + (size_t)(h * BATCH + base) * COND;
  v2f a[8];
  #pragma unroll
  for (int kk = 0; kk < 8; ++kk)
    a[kk] = *(const v2f*)(condH + c16 * COND + kk * 4 + half * 2);

  __syncthreads();

  float O0[8] = {0.f, 0.f, 0.f, 0.f, 0.f, 0.f, 0.f, 0.f};   // o in [0,16)
  float O1[8] = {0.f, 0.f, 0.f, 0.f, 0.f, 0.f, 0.f, 0.f};   // o in [16,32)

  #pragma unroll 1
  for (int i = 0; i < KPI; ++i) {
    // per-lane-half broadcast scalars x1[m][i] for m = 8*half + r
    float xv[8];
    #pragma unroll
    for (int r = 0; r < 8; ++r)
      xv[r] = x1s[wib][(half * 8 + r) * KPI + i];

    const float* Wi = &wLds[i * OUT_F * OSTR];

    { // o-half 0
      v8f acc = {};
      #pragma unroll
      for (int kk = 0; kk < 8; ++kk) {
        v2f b = *(const v2f*)(Wi + c16 * OSTR + kk * 4 + half * 2);
        acc = __builtin_amdgcn_wmma_f32_16x16x4_f32(
            false, a[kk], false, b, (short)0, acc, false, false);
      }
      float bv = bLds[i * OUT_F + c16];
      #pragma unroll
      for (int r = 0; r < 8; ++r) O0[r] += (acc[r] + bv) * xv[r];
    }
    { // o-half 1
      v8f acc = {};
      #pragma unroll
      for (int kk = 0; kk < 8; ++kk) {
        v2f b = *(const v2f*)(Wi + (16 + c16) * OSTR + kk * 4 + half * 2);
        acc = __builtin_amdgcn_wmma_f32_16x16x4_f32(
            false, a[kk], false, b, (short)0, acc, false, false);
      }
      float bv = bLds[i * OUT_F + 16 + c16];
      #pragma unroll
      for (int r = 0; r < 8; ++r) O1[r] += (acc[r] + bv) * xv[r];
    }
  }

  // ---- store: lane holds (m = r + 8*half, o = c16 / 16+c16); coalesced ----
  float* outH = out + (size_t)(h * BATCH + base) * OUT_F;
  #pragma unroll
  for (int r = 0; r < 8; ++r) {
    int m = half * 8 + r;
    outH[m * OUT_F + c16]      = O0[r];
    outH[m * OUT_F + 16 + c16] = O1[r];
  }
}

// ---------------------------------------------------------------------------
extern "C" void kernel_launch(void* const* d_in, const int* in_sizes, int n_in,
                              void* d_out, int out_size, void* d_ws, size_t ws_size,
                              hipStream_t stream) {
  const float* input = (const float*)d_in[0];
  const float* cond  = (const float*)d_in[1];
  const float* W     = (const float*)d_in[2];
  const float* bias  = (const float*)d_in[3];
  float* out = (float*)d_out;

  float* Wr = (float*)d_ws;            // 139264 floats
  float* br = Wr + WRSZ;               // 4352 floats

  reorder_kernel<<<(WRSZ + 255) / 256, 256, 0, stream>>>(W, bias, Wr, br);

  // 8 heads * 256 blocks/head; 8 waves (one 16-sample tile each) per block
  meta_kernel<<<2048, 256, 0, stream>>>(input, cond, Wr, br, out);
}